// MyGVAE_55456617726630
// MI455X (gfx1250) — compile-verified
//
#include <hip/hip_runtime.h>

#define N_NODES 10000
#define F_IN    256
#define H1_DIM  128
#define H2_DIM  64
#define N_EDGES 320000

#define N_TILES   (N_NODES / 16)   // 625
#define KSTEPS    (H2_DIM / 4)     // 16
#define COL_GRP   5                // col tiles per wave (625 = 5 * 125)
#define N_GRPS    (N_TILES / COL_GRP) // 125

typedef float v2f __attribute__((ext_vector_type(2)));
typedef float v8f __attribute__((ext_vector_type(8)));

// ---------------------------------------------------------------------------
// f32 WMMA: D = A(16x4) x B(4x16) + C  (codegen-confirmed round 1)
// ---------------------------------------------------------------------------
__device__ __forceinline__ v8f wmma_f32_acc(v8f acc, v2f a, v2f b) {
  return __builtin_amdgcn_wmma_f32_16x16x4_f32(
      /*neg_a=*/false, a, /*neg_b=*/false, b,
      /*c_mod=*/(short)0, acc, /*reuse_a=*/false, /*reuse_b=*/false);
}

// C[i0:i0+16, j0:j0+16] = A[i0:, :K] * B[:K, j0:], A lda=K, B row-major ldb.
__global__ void gemm16_wmma(const float* __restrict__ A,
                            const float* __restrict__ B,
                            float* __restrict__ C,
                            int K, int ldb) {
  const int i0 = blockIdx.x * 16;
  const int j0 = blockIdx.y * 16;
  const unsigned lane = threadIdx.x & 31u;
  const unsigned half = lane >> 4;
  const unsigned m    = lane & 15u;

  const float* arow = A + (size_t)(i0 + m) * K + 2u * half;
  const float* bcol = B + (size_t)(2u * half) * ldb + (j0 + m);

  v8f acc = {};
  for (int k = 0; k < K; k += 4) {
    v2f a; a.x = arow[k];               a.y = arow[k + 1];
    v2f b; b.x = bcol[(size_t)k * ldb]; b.y = bcol[(size_t)(k + 1) * ldb];
    acc = wmma_f32_acc(acc, a, b);
  }

  float* crow = C + (size_t)(i0 + 8u * half) * ldb + (j0 + m);
#pragma unroll
  for (int r = 0; r < 8; ++r) crow[(size_t)r * ldb] = acc[r];
}

// ---------------------------------------------------------------------------
// Pack mu (final, bias applied) into WMMA fragment order so the adj kernel's
// fragment loads are wave-contiguous 256B transactions:
//   pk[(tile*16 + kstep)*32 + lane] = { mu[(tile*16 + lane%16)*64 + kbase],
//                                       mu[... + 1] },  kbase = 4*kstep + 2*(lane/16)
// ---------------------------------------------------------------------------
__global__ void k_pack(const float* __restrict__ mu, v2f* __restrict__ pk) {
  int idx = blockIdx.x * 256 + threadIdx.x;           // (t*16 + kk)*32 + lane
  if (idx >= N_TILES * KSTEPS * 32) return;
  int lane = idx & 31;
  int kk   = (idx >> 5) & 15;
  int t    = idx >> 9;
  int m = lane & 15, half = lane >> 4;
  const float* p = mu + (size_t)(t * 16 + m) * H2_DIM + kk * 4 + 2 * half;
  v2f v; v.x = p[0]; v.y = p[1];
  pk[idx] = v;
}

// adj = mu * mu^T via packed fragments.  Each wave: one row tile x 5 col
// tiles (16x80 strip), A fragment reused across 5 WMMAs.  NT b32 stores
// (400MB stream, don't pollute L2).
__global__ void adj_wmma(const v2f* __restrict__ pk, float* __restrict__ adj) {
  const int wave = threadIdx.x >> 5;
  const int lane = threadIdx.x & 31;
  const int ti = blockIdx.x;                 // row tile 0..624
  const int tg = blockIdx.y * 8 + wave;      // col group 0..124
  if (tg >= N_GRPS) return;                  // wave-uniform

  const v2f* ap = pk + (size_t)ti * (KSTEPS * 32) + lane;
  const v2f* bp = pk + (size_t)tg * COL_GRP * (KSTEPS * 32) + lane;

  v8f acc[COL_GRP] = {};
#pragma unroll
  for (int kk = 0; kk < KSTEPS; ++kk) {
    v2f a = ap[kk * 32];
#pragma unroll
    for (int c = 0; c < COL_GRP; ++c) {
      v2f b = bp[(size_t)c * (KSTEPS * 32) + kk * 32];
      acc[c] = wmma_f32_acc(acc[c], a, b);
    }
  }

  const int m = lane & 15, half = lane >> 4;
  float* base = adj + (size_t)(ti * 16 + 8 * half) * N_NODES + tg * (COL_GRP * 16) + m;
#pragma unroll
  for (int c = 0; c < COL_GRP; ++c) {
    float* crow = base + c * 16;
#pragma unroll
    for (int r = 0; r < 8; ++r)
      __builtin_nontemporal_store(acc[c][r], crow + (size_t)r * N_NODES);
  }
}

// ---------------------------------------------------------------------------
// GCN normalization plumbing
// ---------------------------------------------------------------------------
__global__ void k_init_deg(float* deg) {
  int i = blockIdx.x * 256 + threadIdx.x;
  if (i < N_NODES) deg[i] = 1.0f;               // self-loop weight
}

__global__ void k_deg_edges(const long long* __restrict__ col,
                            const float* __restrict__ w, float* deg) {
  int e = blockIdx.x * 256 + threadIdx.x;
  if (e < N_EDGES) atomicAdd(&deg[(int)col[e]], w[e]);
}

__global__ void k_rsqrt(float* deg) {
  int i = blockIdx.x * 256 + threadIdx.x;
  if (i < N_NODES) {
    float d = deg[i];
    deg[i] = (d > 0.0f) ? rsqrtf(d) : 0.0f;     // deg -> dis in place
  }
}

__global__ void k_norm(const long long* __restrict__ row,
                       const long long* __restrict__ col,
                       const float* __restrict__ w,
                       const float* __restrict__ dis,
                       float* __restrict__ norm) {
  int e = blockIdx.x * 256 + threadIdx.x;
  if (e < N_EDGES) norm[e] = dis[(int)row[e]] * w[e] * dis[(int)col[e]];
}

// out[i][f] = dis[i]^2 * hw[i][f]   (self-loop contribution initializes acc)
__global__ void k_self_init(const float* __restrict__ hw,
                            const float* __restrict__ dis,
                            float* __restrict__ out, int F) {
  int idx = blockIdx.x * 256 + threadIdx.x;
  if (idx < N_NODES * F) {
    int i = idx / F;
    float d = dis[i];
    out[idx] = d * d * hw[idx];
  }
}

// h1 scatter: out[col] += norm * hw[row], F = 128
__global__ void k_scatter128(const long long* __restrict__ row,
                             const long long* __restrict__ col,
                             const float* __restrict__ norm,
                             const float* __restrict__ hw,
                             float* __restrict__ out) {
  int t = blockIdx.x * 256 + threadIdx.x;
  int e = t >> 7, f = t & 127;
  if (e < N_EDGES) {
    float v = norm[e] * hw[(size_t)row[e] * H1_DIM + f];
    atomicAdd(&out[(size_t)col[e] * H1_DIM + f], v);
  }
}

// mu/logvar scatter fused, F = 64
__global__ void k_scatter64x2(const long long* __restrict__ row,
                              const long long* __restrict__ col,
                              const float* __restrict__ norm,
                              const float* __restrict__ hw2,
                              const float* __restrict__ hw3,
                              float* __restrict__ mu_out,
                              float* __restrict__ lv_out) {
  int t = blockIdx.x * 256 + threadIdx.x;
  int e = t >> 6, f = t & 63;
  if (e < N_EDGES) {
    size_t src = (size_t)row[e] * H2_DIM + f;
    size_t dst = (size_t)col[e] * H2_DIM + f;
    float nv = norm[e];
    atomicAdd(&mu_out[dst], nv * hw2[src]);
    atomicAdd(&lv_out[dst], nv * hw3[src]);
  }
}

__global__ void k_bias_relu(float* __restrict__ h, const float* __restrict__ b) {
  int idx = blockIdx.x * 256 + threadIdx.x;
  if (idx < N_NODES * H1_DIM) {
    float v = h[idx] + b[idx & (H1_DIM - 1)];
    h[idx] = fmaxf(v, 0.0f);
  }
}

__global__ void k_bias2(float* __restrict__ mu, float* __restrict__ lv,
                        const float* __restrict__ b2,
                        const float* __restrict__ b3) {
  int idx = blockIdx.x * 256 + threadIdx.x;
  if (idx < N_NODES * H2_DIM) {
    int f = idx & (H2_DIM - 1);
    mu[idx] += b2[f];
    lv[idx] += b3[f];
  }
}

// ---------------------------------------------------------------------------
extern "C" void kernel_launch(void* const* d_in, const int* in_sizes, int n_in,
                              void* d_out, int out_size, void* d_ws, size_t ws_size,
                              hipStream_t stream) {
  const float*     x    = (const float*)d_in[0];
  const long long* eidx = (const long long*)d_in[1];   // [2, E] int64
  const float*     ew   = (const float*)d_in[2];
  const float*     W1   = (const float*)d_in[3];
  const float*     b1   = (const float*)d_in[4];
  const float*     W2   = (const float*)d_in[5];
  const float*     b2   = (const float*)d_in[6];
  const float*     W3   = (const float*)d_in[7];
  const float*     b3   = (const float*)d_in[8];

  const long long* row = eidx;            // edge_index[0]
  const long long* col = eidx + N_EDGES;  // edge_index[1]

  float* adj = (float*)d_out;                                   // N*N
  float* mu  = adj + (size_t)N_NODES * N_NODES;                 // N*64
  float* lv  = mu + (size_t)N_NODES * H2_DIM;                   // N*64

  // workspace carve-up (floats; pk offset is 8B-aligned)
  float* ws   = (float*)d_ws;
  float* dis  = ws;                               // N
  float* norm = dis + N_NODES;                    // E
  float* hw1  = norm + N_EDGES;                   // N*128
  float* h1   = hw1 + (size_t)N_NODES * H1_DIM;   // N*128
  float* hw2  = h1 + (size_t)N_NODES * H1_DIM;    // N*64
  float* hw3  = hw2 + (size_t)N_NODES * H2_DIM;   // N*64
  v2f*   pk   = (v2f*)(hw3 + (size_t)N_NODES * H2_DIM); // N*64 floats worth

  const int TPB = 256;
  // 1) deg (self-loops = 1) + edge weights, then dis = rsqrt(deg)
  k_init_deg<<<(N_NODES + TPB - 1) / TPB, TPB, 0, stream>>>(dis);
  k_deg_edges<<<(N_EDGES + TPB - 1) / TPB, TPB, 0, stream>>>(col, ew, dis);
  k_rsqrt<<<(N_NODES + TPB - 1) / TPB, TPB, 0, stream>>>(dis);
  k_norm<<<(N_EDGES + TPB - 1) / TPB, TPB, 0, stream>>>(row, col, ew, dis, norm);

  // 2) hw1 = x @ W1  (10000x256 @ 256x128), WMMA f32
  gemm16_wmma<<<dim3(N_TILES, H1_DIM / 16), 32, 0, stream>>>(x, W1, hw1, F_IN, H1_DIM);

  // 3) h1 = selfloop + scatter + bias + relu
  k_self_init<<<(N_NODES * H1_DIM + TPB - 1) / TPB, TPB, 0, stream>>>(hw1, dis, h1, H1_DIM);
  k_scatter128<<<((size_t)N_EDGES * H1_DIM + TPB - 1) / TPB, TPB, 0, stream>>>(row, col, norm, hw1, h1);
  k_bias_relu<<<(N_NODES * H1_DIM + TPB - 1) / TPB, TPB, 0, stream>>>(h1, b1);

  // 4) hw2 = h1 @ W2, hw3 = h1 @ W3  (10000x128 @ 128x64), WMMA f32
  gemm16_wmma<<<dim3(N_TILES, H2_DIM / 16), 32, 0, stream>>>(h1, W2, hw2, H1_DIM, H2_DIM);
  gemm16_wmma<<<dim3(N_TILES, H2_DIM / 16), 32, 0, stream>>>(h1, W3, hw3, H1_DIM, H2_DIM);

  // 5) mu / logvar: selfloop init directly into d_out, scatter, bias
  k_self_init<<<(N_NODES * H2_DIM + TPB - 1) / TPB, TPB, 0, stream>>>(hw2, dis, mu, H2_DIM);
  k_self_init<<<(N_NODES * H2_DIM + TPB - 1) / TPB, TPB, 0, stream>>>(hw3, dis, lv, H2_DIM);
  k_scatter64x2<<<((size_t)N_EDGES * H2_DIM + TPB - 1) / TPB, TPB, 0, stream>>>(row, col, norm, hw2, hw3, mu, lv);
  k_bias2<<<(N_NODES * H2_DIM + TPB - 1) / TPB, TPB, 0, stream>>>(mu, lv, b2, b3);

  // 6) pack mu into fragment order, then adj = mu @ mu^T (WMMA f32, NT stores)
  k_pack<<<(N_TILES * KSTEPS * 32 + TPB - 1) / TPB, TPB, 0, stream>>>(mu, pk);
  dim3 agrid(N_TILES, (N_GRPS + 7) / 8);
  adj_wmma<<<agrid, 256, 0, stream>>>(pk, adj);
}